// AdditiveModel_63711544869098
// MI455X (gfx1250) — compile-verified
//
#include <hip/hip_runtime.h>

typedef float v2f __attribute__((ext_vector_type(2)));
typedef float v8f __attribute__((ext_vector_type(8)));

#define Y_      16
#define Q_      8
#define P_      32
#define D_      8192
#define QP_     (Q_ * P_)          // 256
#define SLICES_ 8
#define DSLICE_ (D_ / SLICES_)     // 1024 d-values per wave
#define NBLK_   (QP_ * SLICES_)    // 2048 partial blocks

// Stage 1: one wave per (q,p,slice). Computes proj[y] = sum_d lam[y,qp,d]*x[qp,d]
// over its D-slice via V_WMMA_F32_16X16X4_F32 (y -> M dim, x broadcast across N),
// scales by c[y,q]*a[y,q,p], writes 16 weighted partials to ws.
__global__ __launch_bounds__(32)
void additive_wmma_stage1(const float* __restrict__ x,
                          const float* __restrict__ lam,
                          const float* __restrict__ a,
                          const float* __restrict__ c,
                          float* __restrict__ ws) {
    const int qp    = blockIdx.x;      // flattened (q,p): qp = q*P + p
    const int slice = blockIdx.y;
    const int lane  = threadIdx.x;     // wave32
    const int row   = lane & 15;       // M = y row this lane feeds for A
    const int hi    = lane >> 4;       // 0: K=0,1   1: K=2,3

    const size_t d0 = (size_t)slice * DSLICE_;

    // A operand: lane(row,hi) streams lam[row, qp, d + 2*hi .. +1]
    const float* lamp = lam + (size_t)row * ((size_t)QP_ * D_)
                            + (size_t)qp * D_ + d0 + 2 * hi;
    // B operand: broadcast x[qp, d + 2*hi .. +1] to all 16 columns
    const float* xp   = x + (size_t)qp * D_ + d0 + 2 * hi;

    v8f acc = {};
    #pragma unroll 4
    for (int i = 0; i < DSLICE_ / 4; ++i) {
        v2f av = *(const v2f*)lamp;    // global_load_b64 (lam stream, 128 MB total)
        v2f bv = *(const v2f*)xp;      // global_load_b64 (broadcast within half-wave)
        acc = __builtin_amdgcn_wmma_f32_16x16x4_f32(
            /*neg_a=*/false, av, /*neg_b=*/false, bv,
            /*c_mod=*/(short)0, acc, /*reuse_a=*/false, /*reuse_b=*/false);
        lamp += 4;
        xp   += 4;
    }

    // C/D layout: VGPR r, lanes 0-15 -> M=r, lanes 16-31 -> M=8+r.
    // All N columns are identical (B was broadcast), so lane 0 holds proj[0..7]
    // and lane 16 holds proj[8..15] in acc[0..7].
    if (row == 0) {
        const int q   = qp / P_;
        float* dst = ws + ((size_t)slice * QP_ + qp) * Y_ + hi * 8;
        #pragma unroll
        for (int r = 0; r < 8; ++r) {
            const int y = hi * 8 + r;
            const float w = c[y * Q_ + q] * a[(size_t)y * QP_ + qp];
            dst[r] = w * acc[r];
        }
    }
}

// Stage 2: deterministic fixed-order reduction of 2048 partial 16-vectors.
__global__ __launch_bounds__(32)
void additive_wmma_stage2(const float* __restrict__ ws, float* __restrict__ out) {
    const int t = threadIdx.x;
    if (t < Y_) {
        float s = 0.0f;
        for (int b = 0; b < NBLK_; ++b) {
            s += ws[(size_t)b * Y_ + t];   // coalesced across the 16 lanes
        }
        out[t] = s;
    }
}

extern "C" void kernel_launch(void* const* d_in, const int* in_sizes, int n_in,
                              void* d_out, int out_size, void* d_ws, size_t ws_size,
                              hipStream_t stream) {
    const float* x   = (const float*)d_in[0];   // (Q,P,D)
    const float* lam = (const float*)d_in[1];   // (Y,Q,P,D)
    const float* a   = (const float*)d_in[2];   // (Y,Q,P)
    const float* c   = (const float*)d_in[3];   // (Y,Q)
    float*       out = (float*)d_out;           // (Y)
    float*       ws  = (float*)d_ws;            // NBLK_*Y_ floats = 128 KB

    dim3 grid1(QP_, SLICES_);
    additive_wmma_stage1<<<grid1, 32, 0, stream>>>(x, lam, a, c, ws);
    additive_wmma_stage2<<<1, 32, 0, stream>>>(ws, out);
}